// UnifiedSASv6_9612136808516
// MI455X (gfx1250) — compile-verified
//
#include <hip/hip_runtime.h>

typedef float v2f __attribute__((ext_vector_type(2)));
typedef float v8f __attribute__((ext_vector_type(8)));

#define IMG_H 512
#define IMG_W 512
#define PLANE_SZ (IMG_H * IMG_W)
#define N_PLANES 96                       // 32 images * 3 channels
#define N_PAIRS  (N_PLANES * 64 * 32)     // 8x16 tile pairs of 8x8 blocks
#define DCT_THRESH 2.5f
#define DCT_CLIP   10.0f
#define DWT_LAM    0.05f

// ds_swizzle_b32 group-of-32 XOR patterns: offset = xor<<10 | or<<5 | and
#define SWZ_XOR1  0x041F
#define SWZ_XOR8  0x201F
#define SWZ_XOR16 0x401F
#define SWZ_XOR24 0x601F

template <int PAT>
__device__ __forceinline__ float swz(float v) {
  return __int_as_float(__builtin_amdgcn_ds_swizzle(__float_as_int(v), PAT));
}

// cos(t*pi/4) table; sin(t*pi/4) = tab[(t+6)&7]
__constant__ float c8tab[8] = {
    1.0f, 0.70710678118654752f, 0.0f, -0.70710678118654752f,
   -1.0f, -0.70710678118654752f, 0.0f, 0.70710678118654752f};

// U = [C; S] (16x8): C[j][k]=cos(2pi jk/8)/sqrt8, S[j][k]=sin(2pi jk/8)/sqrt8
__device__ __forceinline__ float u_entry(int j, int k) {
  const float r8 = 0.35355339059327373f; // 1/sqrt(8)
  int idx = (j < 8) ? ((j * k) & 7) : (((j - 8) * k + 6) & 7);
  return c8tab[idx] * r8;
}

__device__ __forceinline__ v8f wmma_acc(v2f a, v2f b, v8f c) {
  // D(16x16,f32) = A(16x4,f32) * B(4x16,f32) + C
  return __builtin_amdgcn_wmma_f32_16x16x4_f32(
      false, a, false, b, (short)0, c, false, false);
}

// Given A = [X1;X2] (16x8) in A-layout regs (a01 = K0..3 pair, a45 = K4..7 pair)
// and U constants (uk01/uk45 serve both as B=U^T and A=U layouts),
// produce Y1 = U*X1*U^T and Y2 = U*X2*U^T (16x16 each, D-layout).
__device__ __forceinline__ void transform_pair(v2f a01, v2f a45, v2f uk01, v2f uk45,
                                               bool hi, v8f& y1, v8f& y2) {
  v8f w = {};
  w = wmma_acc(a01, uk01, w);   // K = 0..3
  w = wmma_acc(a45, uk45, w);   // K = 4..7  -> W = [X1;X2] * U^T
  float s[8];
#pragma unroll
  for (int r = 0; r < 8; ++r) s[r] = swz<SWZ_XOR16>(w[r]);
  // Y1: B = W[0:8,:]   (D-layout -> B-layout: swap 16-lane halves where needed)
  v2f b0, b1;
  b0[0] = hi ? s[2] : w[0];  b0[1] = hi ? s[3] : w[1];
  b1[0] = hi ? s[6] : w[4];  b1[1] = hi ? s[7] : w[5];
  v8f t = {};
  t = wmma_acc(uk01, b0, t);
  t = wmma_acc(uk45, b1, t);
  y1 = t;
  // Y2: B = W[8:16,:]
  b0[0] = hi ? w[2] : s[0];  b0[1] = hi ? w[3] : s[1];
  b1[0] = hi ? w[6] : s[4];  b1[1] = hi ? w[7] : s[5];
  v8f t2 = {};
  t2 = wmma_acc(uk01, b0, t2);
  t2 = wmma_acc(uk45, b1, t2);
  y2 = t2;
}

__device__ __forceinline__ float dct_norm(float t) {
  float ca = fabsf(t);
  float ts = t * (ca * (1.0f / (DCT_THRESH + 1e-8f)));
  t = (ca < DCT_THRESH) ? ts : t;
  return fminf(fmaxf(t, -DCT_CLIP), DCT_CLIP);
}

__device__ __forceinline__ float softt(float v) {
  float m = fmaxf(fabsf(v) - DWT_LAM, 0.0f);
  return copysignf(m, v);
}

// Haar refine for one 2x2 row pair: u = even row, v = odd row, lane parity = col parity
__device__ __forceinline__ void wavelet_pair(float u, float v, bool even,
                                             float& o0, float& o1) {
  float u2 = swz<SWZ_XOR1>(u);
  float v2 = swz<SWZ_XOR1>(v);
  float a = even ? u : u2, b = even ? u2 : u;
  float c = even ? v : v2, d = even ? v2 : v;
  float LL = 0.5f * (a + b + c + d);
  float LH = softt(0.5f * (a + b - c - d));
  float HL = softt(0.5f * (a - b + c - d));
  float HH = softt(0.5f * (a - b - c + d));
  o0 = even ? 0.5f * (LL + LH + HL + HH) : 0.5f * (LL + LH - HL - HH);
  o1 = even ? 0.5f * (LL - LH + HL - HH) : 0.5f * (LL - LH - HL + HH);
}

// Kernel 1: 3x3 Gaussian blur (reflect pad), staged into out channels 3..5
__global__ void __launch_bounds__(256)
gauss_blur_kernel(const float* __restrict__ in, float* __restrict__ out) {
  const int idx = blockIdx.x * 256 + threadIdx.x;
  const int x = idx & (IMG_W - 1);
  const int y = (idx >> 9) & (IMG_H - 1);
  const int p = idx >> 18;
  const int img = p / 3;
  const int ch  = p - 3 * img;
  const float e1  = expf(-0.78125f);                 // exp(-1/(2*sigma^2)), folds at compile time
  const float inv = 1.0f / ((1.0f + 2.0f * e1) * (1.0f + 2.0f * e1));
  const float* ip = in + (size_t)p * PLANE_SZ;
  const int xm = (x == 0) ? 1 : x - 1;
  const int xp = (x == IMG_W - 1) ? IMG_W - 2 : x + 1;
  const int ym = (y == 0) ? 1 : y - 1;
  const int yp = (y == IMG_H - 1) ? IMG_H - 2 : y + 1;
  const float* r0 = ip + (size_t)ym * IMG_W;
  const float* r1 = ip + (size_t)y  * IMG_W;
  const float* r2 = ip + (size_t)yp * IMG_W;
  float h0 = e1 * r0[xm] + r0[x] + e1 * r0[xp];
  float h1 = e1 * r1[xm] + r1[x] + e1 * r1[xp];
  float h2 = e1 * r2[xm] + r2[x] + e1 * r2[xp];
  float v = inv * (e1 * h0 + h1 + e1 * h2);
  out[(size_t)(img * 6 + 3 + ch) * PLANE_SZ + (size_t)y * IMG_W + x] = v;
}

// Kernel 2: per wave = two horizontally adjacent 8x8 blocks (8x16 tile).
// DCT-normalize via fp32 WMMA, residual, Haar refine, final stores.
__global__ void __launch_bounds__(256)
sas_wmma_kernel(float* __restrict__ out) {
  __shared__ float s_x[8][128];   // staged blurred input, rows [X1;X2]
  __shared__ float s_c[8][128];   // staged normalized coefficients
  const int lane = threadIdx.x & 31;
  const int wv   = threadIdx.x >> 5;
  const int pair = blockIdx.x * 8 + wv;
  if (pair >= N_PAIRS) return;                    // wave-uniform (grid exact)
  const int plane = pair >> 11;                   // / (64*32)
  const int rem   = pair & 2047;
  const int by8   = (rem >> 5) * 8;
  const int bx16  = (rem & 31) * 16;
  const int img   = plane / 3;
  const int ch    = plane - 3 * img;

  float* planeI = out + (size_t)(img * 6 + ch) * PLANE_SZ;
  float* planeR = out + (size_t)(img * 6 + 3 + ch) * PLANE_SZ; // holds staged I_g

  // ---- load stacked A = [X1;X2] (16x8) in WMMA A-layout --------------------
  const int  m  = lane & 15;
  const bool hi = lane >= 16;
  const int  cb = hi ? 2 : 0;                     // K cols {0,1,4,5} / {2,3,6,7}
  const int  srow = m & 7;
  const int  scol = bx16 + ((m < 8) ? 0 : 8) + cb;
  const float* rp = planeR + (size_t)(by8 + srow) * IMG_W + scol;
  float2 p01 = *(const float2*)rp;
  float2 p45 = *(const float2*)(rp + 4);
  s_x[wv][m * 8 + cb]     = p01.x;
  s_x[wv][m * 8 + cb + 1] = p01.y;
  s_x[wv][m * 8 + cb + 4] = p45.x;
  s_x[wv][m * 8 + cb + 5] = p45.y;

  // per-lane U constants: identical values serve as B=U^T (stage 1) and A=U (stage 2)
  v2f uk01, uk45;
  uk01[0] = u_entry(m, hi ? 2 : 0);
  uk01[1] = u_entry(m, hi ? 3 : 1);
  uk45[0] = u_entry(m, hi ? 6 : 4);
  uk45[1] = u_entry(m, hi ? 7 : 5);

  v2f a01, a45;
  a01[0] = p01.x; a01[1] = p01.y;
  a45[0] = p45.x; a45[1] = p45.y;

  // ---- forward transform: Y = U X U^T, coef = Y[0:8,0:8] - Y[8:16,8:16] ----
  v8f y1, y2;
  transform_pair(a01, a45, uk01, uk45, hi, y1, y2);
#pragma unroll
  for (int r = 0; r < 8; ++r) {
    float t1 = y1[r] - swz<SWZ_XOR24>(y1[r]);     // lanes 0..7 valid
    float t2 = y2[r] - swz<SWZ_XOR24>(y2[r]);
    t1 = dct_norm(t1);
    t2 = dct_norm(t2);
    if (lane < 8) {
      s_c[wv][r * 8 + lane]       = t1;           // stacked [coef1;coef2]
      s_c[wv][(8 + r) * 8 + lane] = t2;
    }
  }
  asm volatile("s_wait_dscnt 0" ::: "memory");    // within-wave LDS RAW fence

  // ---- inverse transform (same operator) -----------------------------------
  v2f ic01, ic45;
  ic01[0] = s_c[wv][m * 8 + cb];
  ic01[1] = s_c[wv][m * 8 + cb + 1];
  ic45[0] = s_c[wv][m * 8 + cb + 4];
  ic45[1] = s_c[wv][m * 8 + cb + 5];
  v8f z1, z2;
  transform_pair(ic01, ic45, uk01, uk45, hi, z1, z2);

  // rec rows merged to 16 lanes (lanes 0..7 = block1 cols, 8..15 = block2 cols)
  float rec[8], res[8];
#pragma unroll
  for (int r = 0; r < 8; ++r) {
    float r1  = z1[r] - swz<SWZ_XOR24>(z1[r]);
    float r2  = z2[r] - swz<SWZ_XOR24>(z2[r]);
    float r2s = swz<SWZ_XOR8>(r2);
    float rm  = (lane < 8) ? r1 : r2s;            // I_d row r, col = lane
    int sr    = (lane < 8) ? r : (8 + r);
    float xo  = s_x[wv][sr * 8 + (lane & 7)];     // blurred input
    rec[r] = rm;
    res[r] = xo - rm;                              // R_d
  }

  // ---- Haar DWT refine (local 2x2) + stores --------------------------------
  const bool even = (lane & 1) == 0;
  float* outI = planeI + (size_t)by8 * IMG_W + bx16 + lane;
  float* outR = planeR + (size_t)by8 * IMG_W + bx16 + lane; // overwrites own staged block only
#pragma unroll
  for (int i = 0; i < 4; ++i) {
    float o0, o1, q0, q1;
    wavelet_pair(rec[2 * i], rec[2 * i + 1], even, o0, o1);
    wavelet_pair(res[2 * i], res[2 * i + 1], even, q0, q1);
    if (lane < 16) {
      outI[(size_t)(2 * i) * IMG_W]     = o0;
      outI[(size_t)(2 * i + 1) * IMG_W] = o1;
      outR[(size_t)(2 * i) * IMG_W]     = q0;
      outR[(size_t)(2 * i + 1) * IMG_W] = q1;
    }
  }
}

extern "C" void kernel_launch(void* const* d_in, const int* in_sizes, int n_in,
                              void* d_out, int out_size, void* d_ws, size_t ws_size,
                              hipStream_t stream) {
  (void)in_sizes; (void)n_in; (void)out_size; (void)d_ws; (void)ws_size;
  const float* I = (const float*)d_in[0];
  float* out = (float*)d_out;

  // Stage blurred I_g into out channels 3..5 (overwritten in-place by kernel 2)
  gauss_blur_kernel<<<(N_PLANES * PLANE_SZ) / 256, 256, 0, stream>>>(I, out);

  // 196608 block-pairs, 8 waves (256 threads) per workgroup -> exact grid
  sas_wmma_kernel<<<N_PAIRS / 8, 256, 0, stream>>>(out);
}